// DistanceNeighborList_10402410791123
// MI455X (gfx1250) — compile-verified
//
#include <hip/hip_runtime.h>

typedef float v2f __attribute__((ext_vector_type(2)));
typedef float v8f __attribute__((ext_vector_type(8)));

#define BB   4
#define NN   2048
#define TILES (NN / 16)          // 128 tiles per dim
#define CUT2 25.0f               // CUTOFF^2

// One wave (32 lanes) computes one 16x16 tile of the pair matrix.
// D = A*B + C via v_wmma_f32_16x16x4_f32 gives |xi - xj|^2 for the tile:
//   A row m   = (xi.x, xi.y, xi.z, 1)
//   B col n   = (-2*xj.x, -2*xj.y, -2*xj.z, |xj|^2)
//   C[m][n]   = |xi|^2
// Minimum-image correction per element: d2 = D + s.(s - 2*dx), s = round(fi-fj)*pbc @ cell.
__global__ __launch_bounds__(256) void nbr_tile_kernel(const float* __restrict__ x,
                                                       const float* __restrict__ cell,
                                                       const unsigned char* __restrict__ pbc,
                                                       float* __restrict__ out)
{
    // Per-wave LDS slab: xi[16*3] | fi[16*3] | xj[16*3] | fj[16*3] | sqi[16] | sqj[16] = 224 floats
    __shared__ float lds[8 * 224];
    const int lane = threadIdx.x & 31;
    const int warp = threadIdx.x >> 5;
    float* W = &lds[warp * 224];

    // Tile decomposition: 4 * 128 * 128 = 65536 tiles, 8 tiles per block.
    const int tile = blockIdx.x * 8 + warp;
    const int b  = tile >> 14;          // / (128*128)
    const int r  = tile & 16383;
    const int i0 = (r >> 7) << 4;       // tile row * 16
    const int j0 = (r & 127) << 4;      // tile col * 16

    // ---- uniform: load cell (rows = lattice vectors) and invert 3x3 (adjugate) ----
    const float c00 = cell[0], c01 = cell[1], c02 = cell[2];
    const float c10 = cell[3], c11 = cell[4], c12 = cell[5];
    const float c20 = cell[6], c21 = cell[7], c22 = cell[8];
    const float A0 = c11 * c22 - c12 * c21;
    const float A1 = c12 * c20 - c10 * c22;
    const float A2 = c10 * c21 - c11 * c20;
    const float rdet = 1.0f / (c00 * A0 + c01 * A1 + c02 * A2);
    const float i00 = A0 * rdet;
    const float i01 = (c02 * c21 - c01 * c22) * rdet;
    const float i02 = (c01 * c12 - c02 * c11) * rdet;
    const float i10 = A1 * rdet;
    const float i11 = (c00 * c22 - c02 * c20) * rdet;
    const float i12 = (c02 * c10 - c00 * c12) * rdet;
    const float i20 = A2 * rdet;
    const float i21 = (c01 * c20 - c00 * c21) * rdet;
    const float i22 = (c00 * c11 - c01 * c10) * rdet;
    const float p0 = pbc[0] ? 1.0f : 0.0f;
    const float p1 = pbc[1] ? 1.0f : 0.0f;
    const float p2 = pbc[2] ? 1.0f : 0.0f;

    // ---- stage: lanes 0-15 load i-atoms, 16-31 load j-atoms; compute frac + |x|^2 ----
    const int  a16 = lane & 15;
    const bool isI = lane < 16;
    const int  atom = (isI ? i0 : j0) + a16;
    const float* xp = x + ((size_t)b * NN + atom) * 3;
    const float px = xp[0], py = xp[1], pz = xp[2];
    const float fxv = px * i00 + py * i10 + pz * i20;   // frac = x @ inv_cell (row-vector)
    const float fyv = px * i01 + py * i11 + pz * i21;
    const float fzv = px * i02 + py * i12 + pz * i22;
    const float sq  = px * px + py * py + pz * pz;
    const int xb = isI ? 0 : 96;
    W[xb + a16 * 3 + 0] = px;
    W[xb + a16 * 3 + 1] = py;
    W[xb + a16 * 3 + 2] = pz;
    W[xb + 48 + a16 * 3 + 0] = fxv;
    W[xb + 48 + a16 * 3 + 1] = fyv;
    W[xb + 48 + a16 * 3 + 2] = fzv;
    W[(isI ? 192 : 208) + a16] = sq;
    __syncthreads();

    // ---- build WMMA operands (branchless, EXEC stays all-ones) ----
    // A layout (16x4 f32): lanes 0-15 hold K=0 (V0), K=1 (V1); lanes 16-31 hold K=2, K=3.
    const bool hi = lane >= 16;
    const int  mbase = hi ? 8 : 0;
    v2f Av, Bv;
    Av.x = hi ? W[a16 * 3 + 2]            : W[a16 * 3 + 0];
    Av.y = hi ? 1.0f                      : W[a16 * 3 + 1];
    Bv.x = hi ? (-2.0f * W[96 + a16 * 3 + 2]) : (-2.0f * W[96 + a16 * 3 + 0]);
    Bv.y = hi ? W[208 + a16]              : (-2.0f * W[96 + a16 * 3 + 1]);
    v8f Cv;
#pragma unroll
    for (int v = 0; v < 8; ++v) Cv[v] = W[192 + mbase + v];  // |xi[m]|^2, m = v + mbase

    v8f D = __builtin_amdgcn_wmma_f32_16x16x4_f32(
        /*neg_a=*/false, Av, /*neg_b=*/false, Bv,
        /*c_mod=*/(short)0, Cv, /*reuse_a=*/false, /*reuse_b=*/false);

    // ---- per-element minimum-image correction + store ----
    const int col = a16;
    const float xjx = W[96 + col * 3 + 0], xjy = W[96 + col * 3 + 1], xjz = W[96 + col * 3 + 2];
    const float fjx = W[144 + col * 3 + 0], fjy = W[144 + col * 3 + 1], fjz = W[144 + col * 3 + 2];
    const size_t distBase = (size_t)b * NN * NN;
    const size_t MASKOFF  = (size_t)BB * NN * NN;

#pragma unroll
    for (int v = 0; v < 8; ++v) {
        const int m = mbase + v;
        const float xix = W[m * 3 + 0], xiy = W[m * 3 + 1], xiz = W[m * 3 + 2];
        const float fix = W[48 + m * 3 + 0], fiy = W[48 + m * 3 + 1], fiz = W[48 + m * 3 + 2];
        // integer image shifts (round-half-even matches jnp.round), masked by pbc
        const float n0 = rintf(fix - fjx) * p0;
        const float n1 = rintf(fiy - fjy) * p1;
        const float n2 = rintf(fiz - fjz) * p2;
        // s = n @ cell
        const float s0 = n0 * c00 + n1 * c10 + n2 * c20;
        const float s1 = n0 * c01 + n1 * c11 + n2 * c21;
        const float s2 = n0 * c02 + n1 * c12 + n2 * c22;
        const float dx0 = xix - xjx, dx1 = xiy - xjy, dx2 = xiz - xjz;
        const float corr = s0 * (s0 - 2.0f * dx0) + s1 * (s1 - 2.0f * dx1) + s2 * (s2 - 2.0f * dx2);
        const float d2 = D[v] + corr;
        const bool within = (d2 < CUT2) && (d2 > 0.0f);
        const float dist = within ? sqrtf(d2) : 0.0f;
        const size_t idx = distBase + (size_t)(i0 + m) * NN + (size_t)(j0 + col);
        out[idx]           = dist;                       // dist output
        out[MASKOFF + idx] = within ? 1.0f : 0.0f;       // mask output (as float)
    }
}

extern "C" void kernel_launch(void* const* d_in, const int* in_sizes, int n_in,
                              void* d_out, int out_size, void* d_ws, size_t ws_size,
                              hipStream_t stream) {
    (void)in_sizes; (void)n_in; (void)out_size; (void)d_ws; (void)ws_size;
    const float* x            = (const float*)d_in[0];         // [B, N, 3] f32
    const float* cell         = (const float*)d_in[1];         // [3, 3] f32
    const unsigned char* pbc  = (const unsigned char*)d_in[2]; // [3] bool
    float* out = (float*)d_out;                                // dist | mask, each [B,N,N]

    const int totalTiles = BB * TILES * TILES;   // 65536
    const int blocks = totalTiles / 8;           // 8 waves (tiles) per 256-thread block
    nbr_tile_kernel<<<blocks, 256, 0, stream>>>(x, cell, pbc, out);
}